// custom_SAT_90263032692743
// MI455X (gfx1250) — compile-verified
//
#include <hip/hip_runtime.h>
#include <math.h>

// Problem constants from the reference: B=4, S=10, M=40, H=256
#define NTOK 1600      // B*S*M tokens
#define HDIM 256       // feature dim

typedef __attribute__((ext_vector_type(2))) float v2f;
typedef __attribute__((ext_vector_type(8))) float v8f;

// ---- fast transcendentals: raw hardware TRANS ops, no OCML fixup paths ----
__device__ __forceinline__ float fast_exp2(float x) {
#if __has_builtin(__builtin_amdgcn_exp2f)
    return __builtin_amdgcn_exp2f(x);   // v_exp_f32 (input provably in-range)
#else
    return exp2f(x);
#endif
}

__device__ __forceinline__ float fast_rcp(float x) {
#if __has_builtin(__builtin_amdgcn_rcpf)
    return __builtin_amdgcn_rcpf(x);    // v_rcp_f32 (operand provably normal)
#else
    return 1.0f / x;
#endif
}

__device__ __forceinline__ float fast_tanh(float x) {
#if __has_builtin(__builtin_amdgcn_tanhf)
    return __builtin_amdgcn_tanhf(x);   // native v_tanh_f32
#else
    // branch-free, overflow-stable: tanh(y) = sign(y)*(1 - 2/(e^{2|y|}+1))
    // e^{2|y|} = exp2(2*log2(e)*|y|); overflow -> inf -> 2/inf = 0 -> 1. Exact limit.
    float t = fast_exp2(2.8853900817779268f * fabsf(x));
    float r = 1.0f - 2.0f * fast_rcp(t + 1.0f);
    return copysignf(r, x);
#endif
}

// ---------------------------------------------------------------------------
// Kernel 1: K = tanh(X@W0 + b0), Q = tanh(X@W1 + b1) via V_WMMA_F32_16X16X4_F32
// One wave32 computes one 16x16 fp32 output tile. Grid covers
// 2 matrices * (1600/16) row-tiles * (256/16) col-tiles = 3200 wave-tiles.
// 4 waves per block -> 800 blocks.
// ---------------------------------------------------------------------------
__global__ __launch_bounds__(128) void dense_tanh_wmma(
    const float* __restrict__ X,     // (NTOK, HDIM)
    const float* __restrict__ W0,    // (HDIM, HDIM)
    const float* __restrict__ b0,    // (HDIM)
    const float* __restrict__ W1,
    const float* __restrict__ b1,
    float* __restrict__ Kout,        // (NTOK, HDIM)
    float* __restrict__ Qout)        // (NTOK, HDIM)
{
    const int lane = threadIdx.x & 31;
    const int waveInBlk = threadIdx.x >> 5;
    const int w = blockIdx.x * 4 + waveInBlk;          // global wave-tile id

    const int tilesPerMat = (NTOK / 16) * (HDIM / 16); // 1600
    const int mat   = w / tilesPerMat;                 // 0 -> W0/K, 1 -> W1/Q
    const int t     = w % tilesPerMat;
    const int tileM = t >> 4;                          // 0..99
    const int tileN = t & 15;                          // 0..15

    const float* __restrict__ W    = mat ? W1 : W0;
    const float* __restrict__ bias = mat ? b1 : b0;
    float* __restrict__ Out        = mat ? Qout : Kout;

    const int half = lane >> 4;        // 0: K-pair {0,1}; 1: K-pair {2,3}
    const int sub  = lane & 15;        // M index (for A) / N index (for B)
    const int row  = tileM * 16 + sub; // A-matrix row owned by this lane
    const int col  = tileN * 16 + sub; // B/C/D column owned by this lane

    const float* __restrict__ xrow = X + row * HDIM;

    v8f acc = {};
    #pragma unroll 8
    for (int k = 0; k < HDIM; k += 4) {
        // A fragment (16x4 fp32): lane holds X[row, k+2*half .. k+2*half+1]
        v2f a = *(const v2f*)(xrow + k + 2 * half);
        // B fragment (4x16 fp32): lane holds W[k+2*half, col], W[k+2*half+1, col]
        v2f b;
        b.x = W[(k + 2 * half + 0) * HDIM + col];
        b.y = W[(k + 2 * half + 1) * HDIM + col];
        // 8-arg form: (neg_a, A, neg_b, B, c_mod, C, reuse_a, reuse_b)
        acc = __builtin_amdgcn_wmma_f32_16x16x4_f32(
            false, a, false, b, (short)0, acc, false, false);
    }

    // Epilogue: bias + tanh, scatter per C/D layout (M = r + 8*half, N = col)
    const float bv = bias[col];
    #pragma unroll
    for (int r = 0; r < 8; ++r) {
        const int Mi = r + 8 * half;
        Out[(tileM * 16 + Mi) * HDIM + col] = fast_tanh(acc[r] + bv);
    }
}

// ---------------------------------------------------------------------------
// Kernel 2: per-token softmax-attend.
//   out[i] = sum_j exp(k_i * q_j) * x[j] / sum_j exp(k_i * q_j)
// tanh bounds |k_i*q_j| < 1, so exp never overflows -> max-subtraction in
// softmax is a mathematical no-op and is skipped. One block per token,
// thread i owns output element i; q/x staged in LDS, read as b128 broadcasts.
// exp(k*q) = exp2((k*log2e)*q): single v_mul + native v_exp_f32 per element.
// ---------------------------------------------------------------------------
__global__ __launch_bounds__(256) void attn_softmax(
    const float* __restrict__ X,     // (NTOK, HDIM)
    const float* __restrict__ Kmat,  // (NTOK, HDIM)
    const float* __restrict__ Qmat,  // (NTOK, HDIM)
    float* __restrict__ out)         // (NTOK, HDIM)
{
    __shared__ float sx[HDIM];
    __shared__ float sq[HDIM];

    const int tok = blockIdx.x;
    const int i   = threadIdx.x;

    sx[i] = X[tok * HDIM + i];
    sq[i] = Qmat[tok * HDIM + i];
    __syncthreads();

    const float kl = Kmat[tok * HDIM + i] * 1.4426950408889634f;

    float num = 0.0f, den = 0.0f;
    #pragma unroll 4
    for (int j = 0; j < HDIM; j += 4) {
        const float4 q4 = *(const float4*)&sq[j];
        const float4 x4 = *(const float4*)&sx[j];
        float e0 = fast_exp2(kl * q4.x);
        float e1 = fast_exp2(kl * q4.y);
        float e2 = fast_exp2(kl * q4.z);
        float e3 = fast_exp2(kl * q4.w);
        num = fmaf(e0, x4.x, num); den += e0;
        num = fmaf(e1, x4.y, num); den += e1;
        num = fmaf(e2, x4.z, num); den += e2;
        num = fmaf(e3, x4.w, num); den += e3;
    }
    // den >= 256*e^-1 > 0: raw v_rcp_f32 is safe
    out[tok * HDIM + i] = num * fast_rcp(den);
}

// ---------------------------------------------------------------------------
extern "C" void kernel_launch(void* const* d_in, const int* in_sizes, int n_in,
                              void* d_out, int out_size, void* d_ws, size_t ws_size,
                              hipStream_t stream) {
    const float* X  = (const float*)d_in[0];  // (4,10,40,256) fp32
    const float* W0 = (const float*)d_in[1];  // (256,256)
    const float* b0 = (const float*)d_in[2];  // (256,)
    const float* W1 = (const float*)d_in[3];  // (256,256)
    const float* b1 = (const float*)d_in[4];  // (256,)
    float* out = (float*)d_out;               // (4,10,40,256)

    float* Kws = (float*)d_ws;                       // NTOK*HDIM fp32
    float* Qws = Kws + (size_t)NTOK * HDIM;          // NTOK*HDIM fp32

    // 3200 wave-tiles, 4 waves (128 threads) per block -> 800 blocks
    dense_tanh_wmma<<<800, 128, 0, stream>>>(X, W0, b0, W1, b1, Kws, Qws);

    // One block per token
    attn_softmax<<<NTOK, 256, 0, stream>>>(X, Kws, Qws, out);
}